// DistributedQCompilerGNN_66640712565105
// MI455X (gfx1250) — compile-verified
//
#include <hip/hip_runtime.h>
#include <math.h>

typedef float v2f __attribute__((ext_vector_type(2)));
typedef float v8f __attribute__((ext_vector_type(8)));

// ---------------------------------------------------------------- utilities

__global__ void k_fill(float* __restrict__ p, float v, int n) {
  int i = blockIdx.x * blockDim.x + threadIdx.x;
  if (i < n) p[i] = v;
}

__device__ __forceinline__ void atomicMaxF32(float* addr, float val) {
  int iv = __float_as_int(val);
  if (iv >= 0) atomicMax((int*)addr, iv);
  else         atomicMin((unsigned int*)addr, (unsigned int)iv);
}

// ------------------------------------------------------------ encoder GEMMs
// O[N,64] = relu(X[N,Kin] @ W[64,Kin]^T + b) ; Kin in {1,2,5} -> plain VALU

__global__ void k_encoder(const float* __restrict__ X, const float* __restrict__ W,
                          const float* __restrict__ b, float* __restrict__ O,
                          int N, int Kin) {
  int i = blockIdx.x * blockDim.x + threadIdx.x;
  if (i >= N * 64) return;
  int n = i >> 6, c = i & 63;
  float s = b[c];
  const float* x = X + (size_t)n * Kin;
  const float* w = W + (size_t)c * Kin;
  for (int k = 0; k < Kin; ++k) s = fmaf(x[k], w[k], s);
  O[i] = fmaxf(s, 0.f);
}

// ------------------------------------------------- fp32 WMMA feature GEMM
// Y[N,256] = X[N,64] @ W[256,64]^T + bias.
// One wave computes a 16x64 output strip (4 accumulators): A fragments are
// reused across 4 WMMAs per K-step, cutting X traffic 4x vs 16x16 tiling.
// grid = (ceil(N/16), 4), block = 32 (wave32, EXEC all ones for WMMA).

__global__ void k_gemm_xW(const float* __restrict__ X, const float* __restrict__ W,
                          const float* __restrict__ bias, float* __restrict__ Y,
                          int N) {
  const int lane = threadIdx.x;
  const int half = lane >> 4;     // 0: K{0,1}, 1: K{2,3} within each 4-step
  const int l16  = lane & 15;
  const int tm   = blockIdx.x << 4;
  const int tn   = blockIdx.y << 6;                      // 64 cols per wave
  int arow = tm + l16; if (arow > N - 1) arow = N - 1;   // clamp, keep EXEC full
  const float* xr = X + (size_t)arow * 64;
  const float* w0 = W + (size_t)(tn + l16) * 64;         // B col j*16+l16 = W row
  const float* w1 = w0 + 16 * 64;
  const float* w2 = w0 + 32 * 64;
  const float* w3 = w0 + 48 * 64;
  v8f acc0 = {0.f,0.f,0.f,0.f,0.f,0.f,0.f,0.f};
  v8f acc1 = acc0, acc2 = acc0, acc3 = acc0;
#pragma unroll
  for (int k = 0; k < 64; k += 4) {
    const int kk = k + (half << 1);
    v2f a, b0, b1, b2, b3;
    a.x  = xr[kk]; a.y  = xr[kk + 1];
    b0.x = w0[kk]; b0.y = w0[kk + 1];
    b1.x = w1[kk]; b1.y = w1[kk + 1];
    b2.x = w2[kk]; b2.y = w2[kk + 1];
    b3.x = w3[kk]; b3.y = w3[kk + 1];
    acc0 = __builtin_amdgcn_wmma_f32_16x16x4_f32(false, a, false, b0, (short)0, acc0, false, false);
    acc1 = __builtin_amdgcn_wmma_f32_16x16x4_f32(false, a, false, b1, (short)0, acc1, false, false);
    acc2 = __builtin_amdgcn_wmma_f32_16x16x4_f32(false, a, false, b2, (short)0, acc2, false, false);
    acc3 = __builtin_amdgcn_wmma_f32_16x16x4_f32(false, a, false, b3, (short)0, acc3, false, false);
  }
  const float bc0 = bias[tn + l16];
  const float bc1 = bias[tn + 16 + l16];
  const float bc2 = bias[tn + 32 + l16];
  const float bc3 = bias[tn + 48 + l16];
#pragma unroll
  for (int r = 0; r < 8; ++r) {
    int row = tm + r + (half << 3);
    if (row < N) {
      float* yr = Y + (size_t)row * 256 + tn + l16;
      yr[0]  = acc0[r] + bc0;
      yr[16] = acc1[r] + bc1;
      yr[32] = acc2[r] + bc2;
      yr[48] = acc3[r] + bc3;
    }
  }
}

// --------------------------------------------------------------- edge pass 1
// warp per edge: logits[e,h] = att[h,:] . leakyrelu(xl[src]+xr[dst]); seg-max.

__global__ void k_edge_logits(const float* __restrict__ xl, const float* __restrict__ xr,
                              const int* __restrict__ src, const int* __restrict__ dst,
                              int E, int Nloop, const float* __restrict__ att,
                              float* __restrict__ elog, float* __restrict__ m) {
  int e    = blockIdx.x * (blockDim.x >> 5) + (threadIdx.x >> 5);
  int lane = threadIdx.x & 31;
  int ET   = E + Nloop;
  if (e >= ET) return;
  int s = (e < E) ? src[e] : (e - E);
  int d = (e < E) ? dst[e] : (e - E);
  const float* pl = xl + (size_t)s * 256 + lane * 8;
  const float* pr = xr + (size_t)d * 256 + lane * 8;
  const float* pa = att + lane * 8;
  float partial = 0.f;
#pragma unroll
  for (int j = 0; j < 8; ++j) {
    float h = pl[j] + pr[j];
    h = (h > 0.f) ? h : 0.2f * h;
    partial = fmaf(pa[j], h, partial);
  }
  partial += __shfl_xor(partial, 1, 32);
  partial += __shfl_xor(partial, 2, 32);
  partial += __shfl_xor(partial, 4, 32);
  if ((lane & 7) == 0) {
    int h = lane >> 3;
    elog[(size_t)e * 4 + h] = partial;
    atomicMaxF32(&m[(size_t)d * 4 + h], partial);
  }
}

// --------------------------------------------------------------- edge pass 2

__global__ void k_edge_exp(const float* __restrict__ elog, const int* __restrict__ dst,
                           int E, int Nloop, const float* __restrict__ m,
                           float* __restrict__ eex, float* __restrict__ den) {
  int i = blockIdx.x * blockDim.x + threadIdx.x;
  int ET = E + Nloop;
  if (i >= ET * 4) return;
  int e = i >> 2, h = i & 3;
  int d = (e < E) ? dst[e] : (e - E);
  float ex = expf(elog[i] - m[(size_t)d * 4 + h]);
  eex[i] = ex;
  atomicAdd(&den[(size_t)d * 4 + h], ex);
}

// --------------------------------------------------------------- edge pass 3
// warp per edge, 2 channels/lane; head-mean folded (alpha/H) -> acc[Nd,64]

__global__ void k_edge_aggr(const float* __restrict__ xl, const int* __restrict__ src,
                            const int* __restrict__ dst, int E, int Nloop,
                            const float* __restrict__ eex, const float* __restrict__ den,
                            float* __restrict__ acc) {
  int e    = blockIdx.x * (blockDim.x >> 5) + (threadIdx.x >> 5);
  int lane = threadIdx.x & 31;
  int ET   = E + Nloop;
  if (e >= ET) return;
  int s = (e < E) ? src[e] : (e - E);
  int d = (e < E) ? dst[e] : (e - E);
  const float* pe = eex + (size_t)e * 4;
  const float* pd = den + (size_t)d * 4;
  float a0 = 0.25f * pe[0] / (pd[0] + 1e-16f);
  float a1 = 0.25f * pe[1] / (pd[1] + 1e-16f);
  float a2 = 0.25f * pe[2] / (pd[2] + 1e-16f);
  float a3 = 0.25f * pe[3] / (pd[3] + 1e-16f);
  const float* p = xl + (size_t)s * 256;
  int c = lane * 2;
#pragma unroll
  for (int j = 0; j < 2; ++j) {
    float v = a0 * p[c + j] + a1 * p[64 + c + j] + a2 * p[128 + c + j] + a3 * p[192 + c + j];
    atomicAdd(&acc[(size_t)d * 64 + c + j], v);
  }
}

// ------------------------------------------------- per-layer bias + relu

__global__ void k_finalize(float* __restrict__ acc, const float* __restrict__ b0,
                           const float* __restrict__ b1, const float* __restrict__ b2,
                           int N) {
  int i = blockIdx.x * blockDim.x + threadIdx.x;
  if (i >= N * 64) return;
  int c = i & 63;
  float s = acc[i] + b0[c];
  if (b1) s += b1[c];
  if (b2) s += b2[c];
  acc[i] = fmaxf(s, 0.f);
}

// ------------------------------------------------- attentional pooling

__global__ void k_pool_gate(const float* __restrict__ h, const float* __restrict__ Wg,
                            const float* __restrict__ bg, const int* __restrict__ batch,
                            float* __restrict__ gate, float* __restrict__ gm, int Np) {
  int n = blockIdx.x * blockDim.x + threadIdx.x;
  if (n >= Np) return;
  float s = bg[0];
  const float* p = h + (size_t)n * 64;
  for (int c = 0; c < 64; ++c) s = fmaf(p[c], Wg[c], s);
  gate[n] = s;
  atomicMaxF32(&gm[batch[n]], s);
}

__global__ void k_pool_exp(const float* __restrict__ gate, const int* __restrict__ batch,
                           const float* __restrict__ gm, float* __restrict__ ga,
                           float* __restrict__ gden, int Np) {
  int n = blockIdx.x * blockDim.x + threadIdx.x;
  if (n >= Np) return;
  int b = batch[n];
  float ex = expf(gate[n] - gm[b]);
  ga[n] = ex;
  atomicAdd(&gden[b], ex);
}

__global__ void k_pool_emb(const float* __restrict__ h, const int* __restrict__ batch,
                           const float* __restrict__ ga, const float* __restrict__ gden,
                           float* __restrict__ emb, int Np) {
  int n    = blockIdx.x * (blockDim.x >> 5) + (threadIdx.x >> 5);
  int lane = threadIdx.x & 31;
  if (n >= Np) return;
  int b = batch[n];
  float coef = ga[n] / (gden[b] + 1e-16f);
  const float* p = h + (size_t)n * 64;
  int c = lane * 2;
  atomicAdd(&emb[(size_t)b * 64 + c], coef * p[c]);
  atomicAdd(&emb[(size_t)b * 64 + c + 1], coef * p[c + 1]);
}

// ------------------------------------------------- small head MLP GEMMs

__global__ void k_mlp64(const float* __restrict__ in, const float* __restrict__ W,
                        const float* __restrict__ b, float* __restrict__ out,
                        int Gn, int M, int relu) {
  int i = blockIdx.x * blockDim.x + threadIdx.x;
  if (i >= Gn * M) return;
  int g = i / M, mo = i % M;
  float s = b[mo];
  const float* p = in + (size_t)g * 64;
  const float* w = W + (size_t)mo * 64;
  for (int k = 0; k < 64; ++k) s = fmaf(p[k], w[k], s);
  out[i] = relu ? fmaxf(s, 0.f) : s;
}

// ---------------------------------------------------------------- host side

static inline int cdiv(int a, int b) { return (a + b - 1) / b; }

static void run_gat(hipStream_t s, const float* hsrc, int Ns, const float* hdst, int Nd,
                    const int* ei, int E, bool loops,
                    const float* Wl, const float* bl, const float* Wr, const float* br,
                    const float* att, float* acc,
                    float* xl, float* xr, float* elog, float* eex, float* m, float* den) {
  k_gemm_xW<<<dim3(cdiv(Ns, 16), 4), 32, 0, s>>>(hsrc, Wl, bl, xl, Ns);
  k_gemm_xW<<<dim3(cdiv(Nd, 16), 4), 32, 0, s>>>(hdst, Wr, br, xr, Nd);
  int nm = Nd * 4;
  k_fill<<<cdiv(nm, 256), 256, 0, s>>>(m, -__builtin_inff(), nm);
  k_fill<<<cdiv(nm, 256), 256, 0, s>>>(den, 0.f, nm);
  int Nloop = loops ? Nd : 0;
  int ET = E + Nloop;
  const int* src = ei;
  const int* dst = ei + E;
  k_edge_logits<<<cdiv(ET, 8), 256, 0, s>>>(xl, xr, src, dst, E, Nloop, att, elog, m);
  k_edge_exp<<<cdiv(ET * 4, 256), 256, 0, s>>>(elog, dst, E, Nloop, m, eex, den);
  k_edge_aggr<<<cdiv(ET, 8), 256, 0, s>>>(xl, src, dst, E, Nloop, eex, den, acc);
}

extern "C" void kernel_launch(void* const* d_in, const int* in_sizes, int n_in,
                              void* d_out, int out_size, void* d_ws, size_t ws_size,
                              hipStream_t stream) {
  // ---- inputs (recursive insertion-order flatten of setup_inputs dict) ----
  const float* x_phys = (const float*)d_in[0];
  const float* x_log  = (const float*)d_in[1];
  const float* x_gate = (const float*)d_in[2];
  const int* ei_intra = (const int*)d_in[3];
  const int* ei_inter = (const int*)d_in[4];
  const int* ei_map   = (const int*)d_in[5];
  const int* ei_part  = (const int*)d_in[6];
  const int* ei_dep   = (const int*)d_in[7];
  const int* batch    = (const int*)d_in[8];
  const int Np = in_sizes[0] / 5, Nl = in_sizes[1], Ng = in_sizes[2] / 2;
  const int E_intra = in_sizes[3] / 2, E_inter = in_sizes[4] / 2, E_map = in_sizes[5] / 2;
  const int E_part  = in_sizes[6] / 2, E_dep   = in_sizes[7] / 2;

  // params: We_p,be_p,We_l,be_l,We_g,be_g (9..14), conv1 (15..44), conv2 (45..74),
  // Wg,bg,Wa1,ba1,Wa2,ba2,Wc1,bc1,Wc2,bc2 (75..84).
  // conv relation order: intra, inter, map, part, dep; each: Wl,bl,Wr,br,att,bias.
  auto P = [&](int i) { return (const float*)d_in[i]; };
  const int C1 = 15, C2 = 45;
  auto cb = [&](int base, int rel) { return base + rel * 6; };  // param base for relation

  // ---- workspace bump allocator (floats) ----
  float* ws = (float*)d_ws;
  size_t off = 0;
  auto alloc = [&](size_t n) { float* p = ws + off; off += n; return p; };
  int NsMax = Np > Nl ? Np : Nl; if (Ng > NsMax) NsMax = Ng;
  int NdMax = Np > Ng ? Np : Ng;
  int ETmax = E_intra + Np;
  if (E_inter + Np > ETmax) ETmax = E_inter + Np;
  if (E_dep + Ng > ETmax)   ETmax = E_dep + Ng;
  if (E_part > ETmax)       ETmax = E_part;
  if (E_map > ETmax)        ETmax = E_map;

  float* h_phys = alloc((size_t)Np * 64);
  float* h_log  = alloc((size_t)Nl * 64);
  float* h_gate = alloc((size_t)Ng * 64);
  float* accP   = alloc((size_t)Np * 64);  // layer1 phys acc -> h_phys1
  float* accG   = alloc((size_t)Ng * 64);  // layer1 gate acc -> h_gate1
  float* accP2  = alloc((size_t)Np * 64);  // layer2 phys acc -> h_phys2
  float* xl     = alloc((size_t)NsMax * 256);
  float* xr     = alloc((size_t)NdMax * 256);
  float* elog   = alloc((size_t)ETmax * 4);
  float* eex    = alloc((size_t)ETmax * 4);
  float* mbuf   = alloc((size_t)NdMax * 4);
  float* den    = alloc((size_t)NdMax * 4);
  float* gate   = alloc(Np);
  float* ga     = alloc(Np);
  float* gm     = alloc(16);
  float* gden   = alloc(16);
  float* emb    = alloc(16 * 64);
  float* ha     = alloc(16 * 64);
  float* hc     = alloc(16 * 64);
  (void)ws_size; (void)n_in; (void)out_size;

  // ---- encoders ----
  k_encoder<<<cdiv(Np * 64, 256), 256, 0, stream>>>(x_phys, P(9),  P(10), h_phys, Np, 5);
  k_encoder<<<cdiv(Nl * 64, 256), 256, 0, stream>>>(x_log,  P(11), P(12), h_log,  Nl, 1);
  k_encoder<<<cdiv(Ng * 64, 256), 256, 0, stream>>>(x_gate, P(13), P(14), h_gate, Ng, 2);

  // ---- layer 1 ----
  k_fill<<<cdiv(Np * 64, 256), 256, 0, stream>>>(accP, 0.f, Np * 64);
  k_fill<<<cdiv(Ng * 64, 256), 256, 0, stream>>>(accG, 0.f, Ng * 64);
  {
    int b = cb(C1, 0);  // intra: phys->phys, self loops
    run_gat(stream, h_phys, Np, h_phys, Np, ei_intra, E_intra, true,
            P(b), P(b+1), P(b+2), P(b+3), P(b+4), accP, xl, xr, elog, eex, mbuf, den);
    b = cb(C1, 1);      // inter: phys->phys, self loops
    run_gat(stream, h_phys, Np, h_phys, Np, ei_inter, E_inter, true,
            P(b), P(b+1), P(b+2), P(b+3), P(b+4), accP, xl, xr, elog, eex, mbuf, den);
    b = cb(C1, 2);      // map: log->phys, no loops
    run_gat(stream, h_log, Nl, h_phys, Np, ei_map, E_map, false,
            P(b), P(b+1), P(b+2), P(b+3), P(b+4), accP, xl, xr, elog, eex, mbuf, den);
    b = cb(C1, 3);      // part: log->gate, no loops
    run_gat(stream, h_log, Nl, h_gate, Ng, ei_part, E_part, false,
            P(b), P(b+1), P(b+2), P(b+3), P(b+4), accG, xl, xr, elog, eex, mbuf, den);
    b = cb(C1, 4);      // dep: gate->gate, self loops
    run_gat(stream, h_gate, Ng, h_gate, Ng, ei_dep, E_dep, true,
            P(b), P(b+1), P(b+2), P(b+3), P(b+4), accG, xl, xr, elog, eex, mbuf, den);
  }
  k_finalize<<<cdiv(Np * 64, 256), 256, 0, stream>>>(accP, P(cb(C1,0)+5), P(cb(C1,1)+5), P(cb(C1,2)+5), Np);
  k_finalize<<<cdiv(Ng * 64, 256), 256, 0, stream>>>(accG, P(cb(C1,3)+5), P(cb(C1,4)+5), nullptr, Ng);
  // accP == h_phys1, accG == h_gate1 now (relu applied)

  // ---- layer 2 (gate-side convs skipped: og2 is unused by forward) ----
  k_fill<<<cdiv(Np * 64, 256), 256, 0, stream>>>(accP2, 0.f, Np * 64);
  {
    int b = cb(C2, 0);
    run_gat(stream, accP, Np, accP, Np, ei_intra, E_intra, true,
            P(b), P(b+1), P(b+2), P(b+3), P(b+4), accP2, xl, xr, elog, eex, mbuf, den);
    b = cb(C2, 1);
    run_gat(stream, accP, Np, accP, Np, ei_inter, E_inter, true,
            P(b), P(b+1), P(b+2), P(b+3), P(b+4), accP2, xl, xr, elog, eex, mbuf, den);
    b = cb(C2, 2);
    run_gat(stream, h_log, Nl, accP, Np, ei_map, E_map, false,
            P(b), P(b+1), P(b+2), P(b+3), P(b+4), accP2, xl, xr, elog, eex, mbuf, den);
  }
  k_finalize<<<cdiv(Np * 64, 256), 256, 0, stream>>>(accP2, P(cb(C2,0)+5), P(cb(C2,1)+5), P(cb(C2,2)+5), Np);
  // accP2 == h_phys2

  // ---- attentional pooling over phys nodes ----
  k_fill<<<1, 64, 0, stream>>>(gm, -__builtin_inff(), 16);
  k_fill<<<1, 64, 0, stream>>>(gden, 0.f, 16);
  k_fill<<<cdiv(16 * 64, 256), 256, 0, stream>>>(emb, 0.f, 16 * 64);
  k_pool_gate<<<cdiv(Np, 256), 256, 0, stream>>>(accP2, P(75), P(76), batch, gate, gm, Np);
  k_pool_exp<<<cdiv(Np, 256), 256, 0, stream>>>(gate, batch, gm, ga, gden, Np);
  k_pool_emb<<<cdiv(Np, 8), 256, 0, stream>>>(accP2, batch, ga, gden, emb, Np);

  // ---- heads: logits [16,256] then value [16,1] ----
  float* out = (float*)d_out;
  k_mlp64<<<cdiv(16 * 64, 256), 256, 0, stream>>>(emb, P(77), P(78), ha, 16, 64, 1);
  k_mlp64<<<cdiv(16 * 256, 256), 256, 0, stream>>>(ha, P(79), P(80), out, 16, 256, 0);
  k_mlp64<<<cdiv(16 * 64, 256), 256, 0, stream>>>(emb, P(81), P(82), hc, 16, 64, 1);
  k_mlp64<<<1, 32, 0, stream>>>(hc, P(83), P(84), out + 16 * 256, 16, 1, 0);
}